// CrossUpSamp_70059506532653
// MI455X (gfx1250) — compile-verified
//
#include <hip/hip_runtime.h>

// Problem constants (from reference)
#define N_IN   163842
#define N_F    327680
#define N_NEW  491520
#define N_TOT  (N_IN + N_NEW)   // 655362
#define BC     128              // B * Cin = 2 * 64
#define CIN    64
#define COUT   32
#define GN_EPS 1e-5f

typedef __attribute__((ext_vector_type(2))) float v2f;
typedef __attribute__((ext_vector_type(8))) float v8f;

// Async global->LDS copy path (ASYNCcnt-tracked), guarded so absence of the
// builtin (or the host pass) falls back to plain loads without breaking
// compilation. The builtin signature is (global int* src, lds int* dst,
// imm offset, imm cpol) — int-typed pointers, so cast through int.
#if __has_builtin(__builtin_amdgcn_global_load_async_to_lds_b32) && \
    __has_builtin(__builtin_amdgcn_s_wait_asynccnt)
#define USE_ASYNC_LDS 1
typedef __attribute__((address_space(1))) int as1_int;
typedef __attribute__((address_space(3))) int as3_int;
#else
#define USE_ASYNC_LDS 0
#endif

// ---------------------------------------------------------------------------
// Kernel 1: transpose x (B,Cin,N) = (128 planes of N) -> xT (N, 128)
// so that per-node gathers become 512B-contiguous reads (L2-resident).
// ---------------------------------------------------------------------------
__global__ __launch_bounds__(256) void k_transpose(const float* __restrict__ x,
                                                   float* __restrict__ xT) {
  __shared__ float tile[128 * 65];   // pad 65 to avoid bank conflicts
  const int n0  = blockIdx.x * 64;
  const int tid = threadIdx.x;
  const int nl  = tid & 63;
  const int bq  = tid >> 6;          // 0..3
  #pragma unroll 4
  for (int q = 0; q < 32; ++q) {
    int bc = q * 4 + bq;
    int n  = n0 + nl;
    float v = 0.f;
    if (n < N_IN) v = x[(size_t)bc * N_IN + n];   // coalesced along n
    tile[bc * 65 + nl] = v;
  }
  __syncthreads();
  const int bc2 = tid & 127;
  const int nq  = tid >> 7;          // 0..1
  #pragma unroll 4
  for (int r = 0; r < 32; ++r) {
    int nl2 = r * 2 + nq;
    int n   = n0 + nl2;
    if (n < N_IN) xT[(size_t)n * BC + bc2] = tile[bc2 * 65 + nl2];  // coalesced along bc
  }
}

__global__ void k_zero_stats(float* __restrict__ stats) {
  if (threadIdx.x < 128) stats[threadIdx.x] = 0.f;
}

// ---------------------------------------------------------------------------
// Kernel 2: fused mesh-conv. Block = 128 threads (4 waves), tile = 16 vertices.
// Stage 1: per-(bc) thread computes feat[x,lap,v_ew,v_ns] with all gathers from
//          L2-resident xT (face gradients recomputed, never materialized).
// Stage 2: y[32x16] = coeffs[32x256] x feat[256x16] via V_WMMA_F32_16X16X4_F32,
//          one wave per (batch, M-half), K=256 over 4 independent accumulator
//          chains (16 wmma each) to expose matrix-op ILP.
// ---------------------------------------------------------------------------
__global__ __launch_bounds__(128) void k_meshconv_wmma(
    const float* __restrict__ xT,
    const int*   __restrict__ neighbors, const float* __restrict__ lap_w,
    const int*   __restrict__ face_idx,  const float* __restrict__ gw_ew,
    const float* __restrict__ gw_ns,     const int*   __restrict__ f2v_idx,
    const float* __restrict__ f2v_w,     const float* __restrict__ coeffs,
    const float* __restrict__ bias,      float* __restrict__ y_pre) {
  // feat layout: feat[bc][kk*16 + v], row stride 65 floats (bank-conflict free,
  // and the WMMA B-read pattern lanes0-15 -> K=kk, lanes16-31 -> K=kk+2 is
  // conflict-free too).
  __shared__ float feat[BC * 65];
  __shared__ int   s_nb[16 * 6];
  __shared__ float s_lw[16 * 6];
  __shared__ float s_fw[16 * 6];
  __shared__ int   s_fv[16 * 6 * 3];
  __shared__ float s_ge[16 * 6 * 3];
  __shared__ float s_gn[16 * 6 * 3];

  const int tid = threadIdx.x;
  const int n0  = blockIdx.x * 16;
  const bool fullTile = (n0 + 16 <= N_IN);

  // ---- stage per-tile index/weight data into LDS -------------------------
#if USE_ASYNC_LDS
  bool didAsync = false;
  if (fullTile) {
    if (tid < 96) {
      // contiguous 96-element block copies, global -> LDS, ASYNCcnt-tracked
      __builtin_amdgcn_global_load_async_to_lds_b32(
          (as1_int*)(neighbors + (size_t)n0 * 6 + tid), (as3_int*)&s_nb[tid], 0, 0);
      __builtin_amdgcn_global_load_async_to_lds_b32(
          (as1_int*)(lap_w + (size_t)n0 * 6 + tid), (as3_int*)&s_lw[tid], 0, 0);
      __builtin_amdgcn_global_load_async_to_lds_b32(
          (as1_int*)(f2v_w + (size_t)n0 * 6 + tid), (as3_int*)&s_fw[tid], 0, 0);
    }
    didAsync = true;
  }
  if (!didAsync)
#endif
  {
    if (tid < 96) {
      int v = tid / 6, k = tid % 6;
      int n = n0 + v; if (n >= N_IN) n = N_IN - 1;
      s_nb[tid] = neighbors[n * 6 + k];
      s_lw[tid] = lap_w[n * 6 + k];
      s_fw[tid] = f2v_w[n * 6 + k];
    }
  }
  // face-gradient staging is a gather-of-a-gather; keep it synchronous
  for (int e = tid; e < 288; e += 128) {
    int v = e / 18, rem = e % 18, k = rem / 3, j = rem % 3;
    int n = n0 + v; if (n >= N_IN) n = N_IN - 1;
    int f = f2v_idx[n * 6 + k];
    s_fv[e] = face_idx[f * 3 + j];
    s_ge[e] = gw_ew[f * 3 + j];
    s_gn[e] = gw_ns[f * 3 + j];
  }
#if USE_ASYNC_LDS
  __builtin_amdgcn_s_wait_asynccnt(0);
#endif
  __syncthreads();

  // ---- feature/gather stage: one (b,c) plane per thread ------------------
  const int bc = tid;
  for (int v = 0; v < 16; ++v) {
    int n = n0 + v; if (n >= N_IN) n = N_IN - 1;
    if (v + 1 < 16) {
      int np = n0 + v + 1; if (np >= N_IN) np = N_IN - 1;
      __builtin_prefetch(&xT[(size_t)np * BC + bc], 0, 0);  // global_prefetch_b8
    }
    float xs  = xT[(size_t)n * BC + bc];
    float lap = 0.f;
    #pragma unroll
    for (int k = 0; k < 6; ++k) {
      float xn = xT[(size_t)s_nb[v * 6 + k] * BC + bc];
      lap += (xn - xs) * s_lw[v * 6 + k];
    }
    float vew = 0.f, vns = 0.f;
    #pragma unroll
    for (int k = 0; k < 6; ++k) {
      float ge = 0.f, gn = 0.f;
      #pragma unroll
      for (int j = 0; j < 3; ++j) {
        int   e  = (v * 6 + k) * 3 + j;
        float xf = xT[(size_t)s_fv[e] * BC + bc];
        ge += xf * s_ge[e];
        gn += xf * s_gn[e];
      }
      float w = s_fw[v * 6 + k];
      vew += w * ge;
      vns += w * gn;
    }
    float* fr = &feat[bc * 65];
    fr[0 * 16 + v] = xs;
    fr[1 * 16 + v] = lap;
    fr[2 * 16 + v] = vew;
    fr[3 * 16 + v] = vns;
  }
  __syncthreads();

  // ---- GEMM stage: D(16x16) += A(16x4) x B(4x16), f32 WMMA ---------------
  const int wave = tid >> 5;
  const int lane = tid & 31;
  const int bb   = wave >> 1;        // batch
  const int mh   = wave & 1;         // output half (o in [mh*16, mh*16+16))
  const int vcol = lane & 15;        // N column (vertex) / A row within half
  const int hi   = lane >> 4;        // 0: lanes0-15 (K=0/1), 1: lanes16-31 (K=2/3)
  const int kk0  = hi * 2;

  // A layout (16x4 f32): VGPR0 = K{0|2}, VGPR1 = K{1|3}; rows in lanes.
  const float* aPtr  = coeffs + (mh * 16 + vcol) * (CIN * 4) + kk0;  // 8B aligned
  const float* fBase = &feat[(bb * CIN) * 65];

  // 4 independent accumulator chains -> 4-deep WMMA ILP, summed at the end.
  v8f acc0 = {}, acc1 = {}, acc2 = {}, acc3 = {};
  #pragma unroll 2
  for (int kc = 0; kc < CIN; kc += 4) {
    #pragma unroll
    for (int u = 0; u < 4; ++u) {
      const int kcu = kc + u;
      v2f a;
      a.x = aPtr[kcu * 4 + 0];
      a.y = aPtr[kcu * 4 + 1];
      v2f bmat;
      const float* fr = fBase + kcu * 65;
      bmat.x = fr[kk0 * 16 + vcol];            // B: VGPR0 = K{0|2}
      bmat.y = fr[(kk0 + 1) * 16 + vcol];      //    VGPR1 = K{1|3}
      if (u == 0)
        acc0 = __builtin_amdgcn_wmma_f32_16x16x4_f32(false, a, false, bmat,
                                                     (short)0, acc0, false, false);
      else if (u == 1)
        acc1 = __builtin_amdgcn_wmma_f32_16x16x4_f32(false, a, false, bmat,
                                                     (short)0, acc1, false, false);
      else if (u == 2)
        acc2 = __builtin_amdgcn_wmma_f32_16x16x4_f32(false, a, false, bmat,
                                                     (short)0, acc2, false, false);
      else
        acc3 = __builtin_amdgcn_wmma_f32_16x16x4_f32(false, a, false, bmat,
                                                     (short)0, acc3, false, false);
    }
  }
  v8f acc = (acc0 + acc1) + (acc2 + acc3);

  // D layout: VGPR d -> (M = d + hi*8, N = vcol)
  const int n = n0 + vcol;
  if (n < N_IN) {
    #pragma unroll
    for (int d = 0; d < 8; ++d) {
      int o = mh * 16 + d + hi * 8;
      y_pre[(size_t)(bb * COUT + o) * N_IN + n] = acc[d] + bias[o];
    }
  }
}

// ---------------------------------------------------------------------------
// Kernel 3: per-(b,channel) sum / sumsq over N (groups == channels here).
// ---------------------------------------------------------------------------
__global__ __launch_bounds__(256) void k_stats(const float* __restrict__ y_pre,
                                               float* __restrict__ stats) {
  const int p   = blockIdx.x >> 3;   // 0..63 : (b*32 + o)
  const int seg = blockIdx.x & 7;
  const size_t base = (size_t)p * N_IN;
  float s = 0.f, s2 = 0.f;
  for (int n = seg * 256 + (int)threadIdx.x; n < N_IN; n += 2048) {
    float yv = y_pre[base + n];
    s  += yv;
    s2 += yv * yv;
  }
  #pragma unroll
  for (int off = 16; off > 0; off >>= 1) {
    s  += __shfl_down(s, off, 32);
    s2 += __shfl_down(s2, off, 32);
  }
  __shared__ float red[16];
  const int wave = threadIdx.x >> 5, lane = threadIdx.x & 31;
  if (lane == 0) { red[wave] = s; red[8 + wave] = s2; }
  __syncthreads();
  if (threadIdx.x == 0) {
    float ts = 0.f, ts2 = 0.f;
    #pragma unroll
    for (int w = 0; w < 8; ++w) { ts += red[w]; ts2 += red[8 + w]; }
    atomicAdd(&stats[p * 2 + 0], ts);
    atomicAdd(&stats[p * 2 + 1], ts2);
  }
}

__global__ void k_params(const float* __restrict__ stats,
                         const float* __restrict__ gamma,
                         const float* __restrict__ beta,
                         float* __restrict__ params) {
  const int p = threadIdx.x;
  if (p < 64) {
    const int o   = p & 31;
    const float inv_n = 1.0f / (float)N_IN;
    float mu  = stats[p * 2 + 0] * inv_n;
    float var = stats[p * 2 + 1] * inv_n - mu * mu;
    float inv = rsqrtf(var + GN_EPS);
    float sc  = gamma[o] * inv;
    params[p * 2 + 0] = sc;
    params[p * 2 + 1] = beta[o] - mu * sc;
  }
}

// ---------------------------------------------------------------------------
// Kernel 4: normalize + ReLU + write y part of output.
// ---------------------------------------------------------------------------
__global__ __launch_bounds__(256) void k_apply(const float* __restrict__ y_pre,
                                               const float* __restrict__ params,
                                               float* __restrict__ out) {
  size_t idx = (size_t)blockIdx.x * blockDim.x + threadIdx.x;
  const size_t total = (size_t)64 * N_IN;
  if (idx >= total) return;
  int p = (int)(idx / N_IN);
  int n = (int)(idx - (size_t)p * N_IN);
  float v = y_pre[idx] * params[p * 2] + params[p * 2 + 1];
  out[(size_t)p * N_TOT + n] = v > 0.f ? v : 0.f;
}

// ---------------------------------------------------------------------------
// Kernel 5: upsample — gather normalized y (L2-resident) per plane.
// Thread per new vertex m, loop over the 64 (b,o) planes so the up_idx pair
// is loaded once and reused 64x.
// ---------------------------------------------------------------------------
__global__ __launch_bounds__(256) void k_upsample(const int* __restrict__ up_idx,
                                                  float* __restrict__ out) {
  int m = blockIdx.x * blockDim.x + threadIdx.x;
  if (m >= N_NEW) return;
  const int u0 = up_idx[m * 2 + 0];
  const int u1 = up_idx[m * 2 + 1];
  for (int p = 0; p < 64; ++p) {
    const float* pl = out + (size_t)p * N_TOT;
    out[(size_t)p * N_TOT + N_IN + m] = 0.5f * (pl[u0] + pl[u1]);
  }
}

// ---------------------------------------------------------------------------
extern "C" void kernel_launch(void* const* d_in, const int* in_sizes, int n_in,
                              void* d_out, int out_size, void* d_ws, size_t ws_size,
                              hipStream_t stream) {
  const float* x         = (const float*)d_in[0];
  const int*   neighbors = (const int*)  d_in[1];
  const float* lap_w     = (const float*)d_in[2];
  const int*   face_idx  = (const int*)  d_in[3];
  const float* gw_ew     = (const float*)d_in[4];
  const float* gw_ns     = (const float*)d_in[5];
  const int*   f2v_idx   = (const int*)  d_in[6];
  const float* f2v_w     = (const float*)d_in[7];
  const float* coeffs    = (const float*)d_in[8];
  const float* bias      = (const float*)d_in[9];
  const float* gamma     = (const float*)d_in[10];
  const float* beta      = (const float*)d_in[11];
  const int*   up_idx    = (const int*)  d_in[12];
  float* out = (float*)d_out;

  // Workspace layout (floats): xT | y_pre | stats(128) | params(128)
  float* xT     = (float*)d_ws;
  float* y_pre  = xT + (size_t)N_IN * BC;          // 20,971,776 floats
  float* stats  = y_pre + (size_t)64 * N_IN;       // 10,485,888 floats
  float* params = stats + 128;

  k_transpose<<<(N_IN + 63) / 64, 256, 0, stream>>>(x, xT);
  k_zero_stats<<<1, 128, 0, stream>>>(stats);
  k_meshconv_wmma<<<(N_IN + 15) / 16, 128, 0, stream>>>(
      xT, neighbors, lap_w, face_idx, gw_ew, gw_ns, f2v_idx, f2v_w,
      coeffs, bias, y_pre);
  k_stats<<<64 * 8, 256, 0, stream>>>(y_pre, stats);
  k_params<<<1, 64, 0, stream>>>(stats, gamma, beta, params);
  const size_t total = (size_t)64 * N_IN;
  k_apply<<<(int)((total + 255) / 256), 256, 0, stream>>>(y_pre, params, out);
  k_upsample<<<(N_NEW + 255) / 256, 256, 0, stream>>>(up_idx, out);
}